// LSTMAttentionCell_12455405158974
// MI455X (gfx1250) — compile-verified
//
#include <hip/hip_runtime.h>

// Graves LSTM + attention window, persistent kernel for gfx1250 (MI455X).
// B=128 T=800 U=64 C=80 N=512 K=10.
// f32 WMMA (V_WMMA_F32_16X16X4_F32, inline asm with tied accumulator) for the
// recurrent GEMM; weights prepacked so each lane feeds two WMMAs from one b128
// load; grid-wide barrier between time steps.

#define B_   128
#define T_   800
#define U_   64
#define C_   80
#define N_   512
#define K_   10
#define NBLK 32   // 8 batch-tiles x 4 column-groups

#define WH_KC   64                    // 512 / 8 chunk-pairs
#define WX_KC   11                    // 88 (=83 padded) / 8 chunk-pairs
#define WH_ELE  (WH_KC * 2048 * 8)    // packed W_h floats
#define WX_ELE  (WX_KC * 2048 * 8)    // packed W_x floats

typedef __attribute__((ext_vector_type(2))) float v2f;
typedef __attribute__((ext_vector_type(4))) float v4f;
typedef __attribute__((ext_vector_type(8))) float v8f;

// slot <-> k-offset permutation within a K=8 chunk pair (involution):
// {0,1,4,5,2,3,6,7} : slots [4*hi .. 4*hi+3] hold rows {2hi,2hi+1, 4+2hi,4+2hi+1}
__device__ __host__ __forceinline__ int krel(int j) {
    return (j & 1) | ((j & 2) << 1) | ((j & 4) >> 1);
}

// In-place f32 WMMA: D(=C) += A x B.  Tied "+v" accumulator forces one register
// home for the whole K loop (kills the RA mov ping-pong seen with the builtin).
// D->C accumulate chains have no WMMA hazard (only D->A/B does).
__device__ __forceinline__ void wmma_acc(v8f& acc, v2f a, v2f b) {
    asm("v_wmma_f32_16x16x4_f32 %0, %1, %2, %0"
        : "+v"(acc)
        : "v"(a), "v"(b));
}

__device__ __forceinline__ float fsigmoid(float x) {
    return 1.f / (1.f + __expf(-x));
}
__device__ __forceinline__ float ftanh(float x) {
    return 1.f - 2.f / (__expf(2.f * x) + 1.f);
}

__device__ __forceinline__ void grid_sync(unsigned* cnt, unsigned target) {
    __syncthreads();
    if (threadIdx.x == 0) {
        __threadfence();
        __hip_atomic_fetch_add(cnt, 1u, __ATOMIC_RELEASE, __HIP_MEMORY_SCOPE_AGENT);
        while (__hip_atomic_load(cnt, __ATOMIC_ACQUIRE, __HIP_MEMORY_SCOPE_AGENT) < target) {
            __builtin_amdgcn_s_sleep(1);
        }
        __threadfence();
    }
    __syncthreads();
}

// Pack W_h -> Whp[64][2048][8], W_x (zero-padded to 88 rows) -> Wxp[11][2048][8]
__global__ void prepack_weights(const float* __restrict__ Wx, const float* __restrict__ Wh,
                                float* __restrict__ Wxp, float* __restrict__ Whp) {
    int i = blockIdx.x * blockDim.x + threadIdx.x;
    if (i < WH_ELE) {
        int j = i & 7, colc = (i >> 3) & 2047, kc = i >> 14;
        Whp[i] = Wh[(size_t)(kc * 8 + krel(j)) * 2048 + colc];
    } else if (i < WH_ELE + WX_ELE) {
        int ii = i - WH_ELE;
        int j = ii & 7, colc = (ii >> 3) & 2047, kc = ii >> 14;
        int row = kc * 8 + krel(j);
        Wxp[ii] = (row < 83) ? Wx[(size_t)row * 2048 + colc] : 0.f;
    }
}

__global__ void init_ws(float* ws, int nwords) {
    int i = blockIdx.x * blockDim.x + threadIdx.x;
    if (i < nwords) ws[i] = 0.f;
}

__global__ __launch_bounds__(256)
void lstm_attn_persistent(const float* __restrict__ x,      // [B,T,3]
                          const float* __restrict__ sent,   // [B,U,C]
                          const int*   __restrict__ slen,   // [B]
                          const float* __restrict__ Wxp,    // packed [11][2048][8]
                          const float* __restrict__ Whp,    // packed [64][2048][8]
                          const float* __restrict__ bias,   // [2048]
                          const float* __restrict__ Wwin,   // [512,30]
                          const float* __restrict__ bwin,   // [30]
                          float* __restrict__ h_out,        // [B,T,N]
                          float* __restrict__ phi_out,      // [B,T,U]
                          float* __restrict__ w_out,        // [B,T,C]
                          float* __restrict__ h0buf, float* __restrict__ h1buf,
                          float* __restrict__ c_state,
                          float* __restrict__ kappa_state,
                          float* __restrict__ w_state,
                          unsigned* bar)
{
    // LDS: A staging in permuted chunk-pair layout (stride 516 => 16B aligned,
    // bank-conflict-free b128 reads) + attention scratch.
    __shared__ __align__(16) float A_x[16][92];    // 88 packed xi slots + pad
    __shared__ __align__(16) float A_h[16][516];   // 512 packed h slots + pad
    __shared__ float winS[16][32];
    __shared__ float alphaS[16][10], betaS[16][10], kapS[16][10];
    __shared__ float phiS[16][64];

    const int tid      = threadIdx.x;
    const int btile    = blockIdx.x >> 2;   // 0..7  (16 batch rows each)
    const int colgroup = blockIdx.x & 3;    // 0..3  (128 hidden cols each)
    const int wave     = tid >> 5;          // 0..7
    const int lane     = tid & 31;
    const int ln       = lane & 15;         // A row / B,C col within tile
    const int hi       = lane >> 4;
    const int cn       = colgroup * 128 + wave * 16;
    const int col      = cn + ln;           // hidden column this lane owns

    // Per-gate invariants: packed-B element offsets and bias values.
    size_t zoff[4];
    float bb[4];
#pragma unroll
    for (int q = 0; q < 4; ++q) {
        zoff[q] = ((size_t)((q << 9) + col) << 3) + (hi << 2);
        bb[q]   = bias[(q << 9) + col];
    }

    unsigned epoch = 0;

    for (int t = 0; t < T_; ++t) {
        const float* hprev = (t & 1) ? h1buf : h0buf;
        float*       hnext = (t & 1) ? h0buf : h1buf;

        // ---- stage A operands (permuted layout): xi = [x_t | w_prev], h_prev ----
        for (int idx = tid; idx < 16 * 88; idx += 256) {
            int r = idx / 88, k = idx % 88;
            int src = (k & ~7) | krel(k & 7);
            int bg = btile * 16 + r;
            float v = 0.f;
            if (src < 3)       v = x[(size_t)bg * (T_ * 3) + (size_t)t * 3 + src];
            else if (src < 83) v = w_state[bg * C_ + (src - 3)];
            A_x[r][k] = v;
        }
        for (int idx = tid; idx < 16 * 512; idx += 256) {
            int r = idx >> 9, k = idx & 511;
            int src = (k & ~7) | krel(k & 7);
            A_h[r][k] = hprev[(size_t)(btile * 16 + r) * N_ + src];
        }
        __syncthreads();

        // ---- z = xi@Wx + h@Wh + b : one 16x16 tile x 4 gates per wave ----
        v8f acc[4];
#pragma unroll
        for (int q = 0; q < 4; ++q)
#pragma unroll
            for (int v = 0; v < 8; ++v) acc[q][v] = bb[q];

        // xi part: 11 chunk-pairs (K=8 each), zero-padded, no guards
        for (int kc = 0; kc < WX_KC; ++kc) {
            v4f a4 = *(const v4f*)&A_x[ln][(kc << 3) + (hi << 2)];
            v2f a0, a1;
            a0[0] = a4[0]; a0[1] = a4[1];
            a1[0] = a4[2]; a1[1] = a4[3];
            v4f b4[4];
#pragma unroll
            for (int q = 0; q < 4; ++q)
                b4[q] = *(const v4f*)(Wxp + ((size_t)kc << 14) + zoff[q]);
#pragma unroll
            for (int q = 0; q < 4; ++q) {
                v2f b0; b0[0] = b4[q][0]; b0[1] = b4[q][1];
                wmma_acc(acc[q], a0, b0);
            }
#pragma unroll
            for (int q = 0; q < 4; ++q) {
                v2f b1; b1[0] = b4[q][2]; b1[1] = b4[q][3];
                wmma_acc(acc[q], a1, b1);
            }
        }
        // h part: 64 chunk-pairs
#pragma unroll 2
        for (int kc = 0; kc < WH_KC; ++kc) {
            v4f a4 = *(const v4f*)&A_h[ln][(kc << 3) + (hi << 2)];
            v2f a0, a1;
            a0[0] = a4[0]; a0[1] = a4[1];
            a1[0] = a4[2]; a1[1] = a4[3];
            v4f b4[4];
#pragma unroll
            for (int q = 0; q < 4; ++q)
                b4[q] = *(const v4f*)(Whp + ((size_t)kc << 14) + zoff[q]);
#pragma unroll
            for (int q = 0; q < 4; ++q) {
                v2f b0; b0[0] = b4[q][0]; b0[1] = b4[q][1];
                wmma_acc(acc[q], a0, b0);
            }
#pragma unroll
            for (int q = 0; q < 4; ++q) {
                v2f b1; b1[0] = b4[q][2]; b1[1] = b4[q][3];
                wmma_acc(acc[q], a1, b1);
            }
        }

        // ---- gates + cell/hidden update (C/D layout: m = v + 8*hi, n = ln) ----
#pragma unroll
        for (int v = 0; v < 8; ++v) {
            int mrow = v + (hi << 3);
            int bg = btile * 16 + mrow;
            size_t sidx = (size_t)bg * N_ + col;
            float ig = fsigmoid(acc[0][v]);
            float fg = fsigmoid(acc[1][v]);
            float gg = ftanh(acc[2][v]);
            float og = fsigmoid(acc[3][v]);
            float cv = fg * c_state[sidx] + ig * gg;
            c_state[sidx] = cv;
            float hv = og * ftanh(cv);
            hnext[sidx] = hv;
            h_out[(size_t)bg * (T_ * N_) + (size_t)t * N_ + col] = hv;
        }

        grid_sync(bar, (++epoch) * NBLK);

        // ---- attention window: one block (colgroup 0) per batch tile ----
        if (colgroup == 0) {
            // restage h_new (linear layout; phase-2 private use of A_h)
            for (int idx = tid; idx < 16 * 512; idx += 256) {
                int r = idx >> 9, cc = idx & 511;
                A_h[r][cc] = hnext[(size_t)(btile * 16 + r) * N_ + cc];
            }
            __syncthreads();
            // win = h @ W_win + b_win  -> [16, 30]
            for (int idx = tid; idx < 16 * 30; idx += 256) {
                int r = idx / 30, j = idx % 30;
                float s = bwin[j];
                for (int k = 0; k < 512; ++k) s += A_h[r][k] * Wwin[k * 30 + j];
                winS[r][j] = s;
            }
            __syncthreads();
            // alpha, beta, kappa update
            for (int idx = tid; idx < 16 * 10; idx += 256) {
                int r = idx / 10, k = idx % 10;
                int bg = btile * 16 + r;
                alphaS[r][k] = __expf(winS[r][k]);
                betaS[r][k]  = __expf(winS[r][10 + k]);
                float kap = kappa_state[bg * K_ + k] + __expf(winS[r][20 + k]);
                kappa_state[bg * K_ + k] = kap;
                kapS[r][k] = kap;
            }
            __syncthreads();
            // phi[b,u] = sum_k alpha * exp(-beta*(kappa-(u+1))^2), masked
            for (int idx = tid; idx < 16 * 64; idx += 256) {
                int r = idx >> 6, u = idx & 63;
                int bg = btile * 16 + r;
                float uu = (float)(u + 1);
                float s = 0.f;
#pragma unroll
                for (int k = 0; k < 10; ++k) {
                    float d = kapS[r][k] - uu;
                    s += alphaS[r][k] * __expf(-betaS[r][k] * d * d);
                }
                s = (u < slen[bg]) ? s : 0.f;
                phiS[r][u] = s;
                phi_out[(size_t)bg * (T_ * U_) + (size_t)t * U_ + u] = s;
            }
            __syncthreads();
            // w[b,c] = sum_u phi[b,u] * sent[b,u,c]
            for (int idx = tid; idx < 16 * 80; idx += 256) {
                int r = idx / 80, cc = idx % 80;
                int bg = btile * 16 + r;
                const float* sp = sent + (size_t)bg * (U_ * C_) + cc;
                float s = 0.f;
                for (int u = 0; u < 64; ++u) s += phiS[r][u] * sp[u * C_];
                w_state[bg * C_ + cc] = s;
                w_out[(size_t)bg * (T_ * C_) + (size_t)t * C_ + cc] = s;
            }
        }

        grid_sync(bar, (++epoch) * NBLK);
    }
}

extern "C" void kernel_launch(void* const* d_in, const int* in_sizes, int n_in,
                              void* d_out, int out_size, void* d_ws, size_t ws_size,
                              hipStream_t stream) {
    const float* x    = (const float*)d_in[0];
    const float* sent = (const float*)d_in[1];
    const int*   slen = (const int*)d_in[2];
    const float* Wx   = (const float*)d_in[3];
    const float* Wh   = (const float*)d_in[4];
    const float* bias = (const float*)d_in[5];
    const float* Wwin = (const float*)d_in[6];
    const float* bwin = (const float*)d_in[7];

    float* out     = (float*)d_out;
    float* h_out   = out;                                  // [B,T,N]
    float* phi_out = out + (size_t)B_ * T_ * N_;           // [B,T,U]
    float* w_out   = phi_out + (size_t)B_ * T_ * U_;       // [B,T,C]

    // Workspace: Whp | Wxp | h0 | h1 | c | kappa | w | barrier  (~5.8 MB)
    float* ws    = (float*)d_ws;
    float* Whp   = ws;
    float* Wxp   = Whp + WH_ELE;
    float* h0buf = Wxp + WX_ELE;
    float* h1buf = h0buf + (size_t)B_ * N_;
    float* c_st  = h1buf + (size_t)B_ * N_;
    float* kap   = c_st  + (size_t)B_ * N_;
    float* wst   = kap   + (size_t)B_ * K_;
    unsigned* bar = (unsigned*)(wst + (size_t)B_ * C_);

    int nstate = 3 * B_ * N_ + B_ * K_ + B_ * C_ + 1;      // states + barrier word
    init_ws<<<(nstate + 255) / 256, 256, 0, stream>>>(h0buf, nstate);

    int npack = WH_ELE + WX_ELE;
    prepack_weights<<<(npack + 255) / 256, 256, 0, stream>>>(Wx, Wh, Wxp, Whp);

    lstm_attn_persistent<<<NBLK, 256, 0, stream>>>(
        x, sent, slen, Wxp, Whp, bias, Wwin, bwin,
        h_out, phi_out, w_out,
        h0buf, h1buf, c_st, kap, wst, bar);
}